// VertexUpdate_91096256348952
// MI455X (gfx1250) — compile-verified
//
#include <hip/hip_runtime.h>
#include <stdint.h>

// ---------------------------------------------------------------------------
// VertexUpdate (Jacobi-style relaxation step):
//   cbar = segment_sum(edge_attr, dst);  x_new = x + w*(b - cbar)/A
// Memory-bound: ~156 MB traffic -> ~6.7us floor at 23.3 TB/s.
// CDNA5 paths: TENSOR_LOAD_TO_LDS (TDM, double-buffered edge staging),
// s_wait_tensorcnt, ds_load_b64 tile reads, and hardware
// global_atomic_add_f32 scope:SCOPE_DEV (4 MB accumulator is L2-resident).
// ---------------------------------------------------------------------------

#define TILE     4096      // edges per TDM tile (16 KB vals + 16 KB dsts)
#define THREADS  256       // 8 waves (wave32)

typedef unsigned int uint32x4 __attribute__((ext_vector_type(4)));
typedef int          int32x4  __attribute__((ext_vector_type(4)));
typedef int          int32x8  __attribute__((ext_vector_type(8)));
typedef float        f32x2    __attribute__((ext_vector_type(2)));
typedef int          i32x2    __attribute__((ext_vector_type(2)));

// Guaranteed single-instruction FP32 atomic to L2 (no CAS loop).
__device__ __forceinline__ void atomic_add_f32_dev(float* addr, float v) {
    asm volatile("global_atomic_add_f32 %0, %1, off scope:SCOPE_DEV"
                 :: "v"(addr), "v"(v) : "memory");
}

// Two guaranteed LDS loads (vals pair + dsts pair) + one dscnt wait.
__device__ __forceinline__ void lds_load_pair(uint32_t off_v, uint32_t off_d,
                                              f32x2& v, i32x2& d) {
    asm volatile("ds_load_b64 %0, %2\n\t"
                 "ds_load_b64 %1, %3\n\t"
                 "s_wait_dscnt 0x0"
                 : "=&v"(v), "=&v"(d)
                 : "v"(off_v), "v"(off_d)
                 : "memory");
}

#if __has_builtin(__builtin_amdgcn_tensor_load_to_lds)
#define HAVE_TDM 1

__device__ __forceinline__ void tdm_wait() {
#if __has_builtin(__builtin_amdgcn_s_wait_tensorcnt)
    __builtin_amdgcn_s_wait_tensorcnt(0);
#else
    asm volatile("s_wait_tensorcnt 0x0" ::: "memory");
#endif
    asm volatile("" ::: "memory");   // LDS was written behind the optimizer
}

// 1D contiguous DMA: copy `valid_elems` dwords from gptr into LDS at
// lds_byte_off (tile_dim0 = TILE; OOB reads beyond tensor_dim0 return zero).
__device__ __forceinline__ void tdm_load_1d(uint32_t lds_byte_off,
                                            const void* gptr,
                                            uint32_t valid_elems) {
    uint64_t ga = (uint64_t)(uintptr_t)gptr;
    uint32x4 g0;
    g0.x = 1u;                                      // count=1, user descriptor
    g0.y = lds_byte_off;                            // lds_addr (bytes)
    g0.z = (uint32_t)ga;                            // global_addr[31:0]
    g0.w = ((uint32_t)(ga >> 32) & 0x01FFFFFFu)     // global_addr[56:32]
         | (2u << 30);                              // type = 2 ("image")
    int32x8 g1;
    g1[0] = (int)(2u << 16);                        // data_size = 4 bytes
    g1[1] = (int)((valid_elems & 0xFFFFu) << 16);   // tensor_dim0[15:0]
    g1[2] = (int)(((valid_elems >> 16) & 0xFFFFu)   // tensor_dim0[31:16]
         | (1u << 16));                             // tensor_dim1 = 1
    g1[3] = (int)((uint32_t)TILE << 16);            // tile_dim0 = TILE
    g1[4] = 1;                                      // tile_dim1 = 1
    g1[5] = TILE;                                   // tensor_dim0_stride
    g1[6] = 0;
    g1[7] = 0;
    int32x4 z4 = {0, 0, 0, 0};
#if __clang_major__ >= 23
    int32x8 z8 = {0, 0, 0, 0, 0, 0, 0, 0};          // amdgpu-toolchain 6-arg form
    __builtin_amdgcn_tensor_load_to_lds(g0, g1, z4, z4, z8, 0);
#else
    __builtin_amdgcn_tensor_load_to_lds(g0, g1, z4, z4, 0);  // ROCm 7.2 5-arg
#endif
}
#endif // HAVE_TDM

#if defined(HAVE_TDM)
// TDM-staged, double-buffered edge scatter (defined first for the disasm).
__global__ void __launch_bounds__(THREADS)
edge_scatter_kernel(const float* __restrict__ evals,
                    const int*   __restrict__ dsts,
                    float*       __restrict__ cbar,
                    int n_edges, int n_tiles) {
    __shared__ float s_val[2][TILE];
    __shared__ int   s_dst[2][TILE];

    const bool wave0 = (threadIdx.x >> 5) == 0;   // uniform per wave
    int t = (int)blockIdx.x;
    if (t >= n_tiles) return;
    int buf = 0;

    if (wave0) {  // TDM ignores EXEC; gate issue at wave granularity
        const int base = t * TILE;
        const uint32_t valid = (uint32_t)min(TILE, n_edges - base);
        tdm_load_1d((uint32_t)(uintptr_t)&s_val[buf][0], evals + base, valid);
        tdm_load_1d((uint32_t)(uintptr_t)&s_dst[buf][0], dsts  + base, valid);
    }

    while (t < n_tiles) {
        const int t_next = t + (int)gridDim.x;

        if (wave0) tdm_wait();       // DMA for (t, buf) complete
        __syncthreads();             // publish LDS tile to all 8 waves

        if (wave0 && t_next < n_tiles) {   // prefetch next tile into buf^1
            const int base = t_next * TILE;
            const uint32_t valid = (uint32_t)min(TILE, n_edges - base);
            tdm_load_1d((uint32_t)(uintptr_t)&s_val[buf ^ 1][0], evals + base, valid);
            tdm_load_1d((uint32_t)(uintptr_t)&s_dst[buf ^ 1][0], dsts  + base, valid);
        }

        // Process tile t: 2 edges per step via ds_load_b64 pairs.
        const int remain = n_edges - t * TILE;
        #pragma unroll
        for (int p = (int)threadIdx.x; p < TILE / 2; p += THREADS) {  // 8 iters
            const uint32_t off_v = (uint32_t)(uintptr_t)&s_val[buf][2 * p];
            const uint32_t off_d = (uint32_t)(uintptr_t)&s_dst[buf][2 * p];
            f32x2 v; i32x2 d;
            lds_load_pair(off_v, off_d, v, d);
            const int e = 2 * p;
            if (e < remain)     atomic_add_f32_dev(cbar + d.x, v.x);
            if (e + 1 < remain) atomic_add_f32_dev(cbar + d.y, v.y);
        }

        __syncthreads();             // tile consumed before it is refilled
        t = t_next;
        buf ^= 1;
    }
}
#else
// Fallback: direct coalesced streaming + hardware atomics.
__global__ void __launch_bounds__(THREADS)
edge_scatter_kernel(const float* __restrict__ evals,
                    const int*   __restrict__ dsts,
                    float*       __restrict__ cbar,
                    int n_edges, int /*n_tiles*/) {
    int i = blockIdx.x * blockDim.x + threadIdx.x;
    const int stride = gridDim.x * blockDim.x;
    for (; i < n_edges; i += stride)
        atomic_add_f32_dev(cbar + dsts[i], evals[i]);
}
#endif

__global__ void __launch_bounds__(THREADS)
zero_kernel(float* __restrict__ p, int n) {
    int i = blockIdx.x * blockDim.x + threadIdx.x;
    if (i < n) p[i] = 0.0f;
}

__global__ void __launch_bounds__(THREADS)
vertex_kernel(const float* __restrict__ va,
              const float* __restrict__ cbar,
              const float* __restrict__ g,
              float* __restrict__ out, int n) {
    int i = blockIdx.x * blockDim.x + threadIdx.x;
    if (i < n) {
        const float w = g[0];
        const float A = va[3 * i + 0];
        const float b = va[3 * i + 1];
        const float x = va[3 * i + 2];
        const float c = cbar[i];
        out[3 * i + 0] = A;
        out[3 * i + 1] = b;
        out[3 * i + 2] = x + w * (b - c) / A;
    }
}

extern "C" void kernel_launch(void* const* d_in, const int* in_sizes, int n_in,
                              void* d_out, int out_size, void* d_ws, size_t ws_size,
                              hipStream_t stream) {
    // setup_inputs order: vertex_attr, edgeij_pair, edge_attr, g, batch
    const float* vertex_attr = (const float*)d_in[0];
    const int*   edge_pair   = (const int*)  d_in[1];   // (2, E): row 0 = dst
    const float* edge_attr   = (const float*)d_in[2];
    const float* g           = (const float*)d_in[3];

    const int n = in_sizes[0] / 3;       // 1,000,000 vertices
    const int E = in_sizes[2];           // 16,000,000 edges
    const int* dst = edge_pair;          // first E entries = destination row

    float* cbar = (float*)d_ws;          // 4 MB accumulator (L2-resident)
    float* out  = (float*)d_out;

    // 1) Zero the accumulator every call (graph replays don't reset d_ws).
    zero_kernel<<<(n + THREADS - 1) / THREADS, THREADS, 0, stream>>>(cbar, n);

    // 2) Edge scatter-add through TDM-staged LDS tiles.
    const int n_tiles = (E + TILE - 1) / TILE;
    int grid = 1024;
    if (grid > n_tiles) grid = n_tiles;
    edge_scatter_kernel<<<grid, THREADS, 0, stream>>>(edge_attr, dst, cbar, E, n_tiles);

    // 3) Vertex update + output assembly.
    vertex_kernel<<<(n + THREADS - 1) / THREADS, THREADS, 0, stream>>>(
        vertex_attr, cbar, g, out, n);
}